// AEGNet_82119774699540
// MI455X (gfx1250) — compile-verified
//
#include <hip/hip_runtime.h>
#include <stdint.h>

#define BN   128
#define ON   1024
#define IDIM 1024
#define EPSF 1e-7f

__device__ __forceinline__ float fast_rsq(float x) { return __builtin_amdgcn_rsqf(x); }

// ---------------------------------------------------------------------------
// Kernel 0: fold the 6 parameter tensors into T[o,i] and its transpose Tt[i,o]
// via an LDS-tiled 32x32 transpose (both writes coalesced).
// ---------------------------------------------------------------------------
__global__ __launch_bounds__(256) void prep_kernel(
    const float* __restrict__ input_x, const float* __restrict__ input_y,
    const float* __restrict__ output_x, const float* __restrict__ output_y,
    const float* __restrict__ linkage_add, const float* __restrict__ linkage_mul,
    float* __restrict__ T, float* __restrict__ Tt) {
  __shared__ float tile[32][33];  // +1 pad: no LDS bank conflicts on transpose
  const int i0 = blockIdx.x * 32;
  const int o0 = blockIdx.y * 32;
  const int li = threadIdx.x;     // 0..31
  for (int r = threadIdx.y; r < 32; r += 8) {
    const int idx = (o0 + r) * IDIM + (i0 + li);
    const float t = (input_x[idx] / input_y[idx] + linkage_add[idx]) *
                        (1.0f + linkage_mul[idx]) -
                    output_x[idx] / output_y[idx];
    T[idx] = t;
    tile[r][li] = t;
  }
  __syncthreads();
  for (int r = threadIdx.y; r < 32; r += 8) {
    Tt[(i0 + r) * ON + (o0 + li)] = tile[li][r];
  }
}

// ---------------------------------------------------------------------------
// Kernel 1: per (b,i): m = max_o s, Z = sum_o exp(s-m); store {g^2, d/Z, m}.
// Thread-per-i => every T read is a coalesced 1KB row segment; no reductions.
// ---------------------------------------------------------------------------
__global__ __launch_bounds__(256) void pass1_kernel(
    const float* __restrict__ data, const float* __restrict__ T,
    float4* __restrict__ aux) {
  const int b = blockIdx.y;
  const int i = blockIdx.x * 256 + threadIdx.x;
  const float dv = data[b * IDIM + i];
  const float g  = 1.0f / (1.0f + __expf(-dv));  // sigmoid
  const float h  = g * g;
  const float* tp = T + i;

  float m = 0.0f;  // s = rsqrt(positive) > 0, so 0 is a safe identity
#pragma unroll 8
  for (int o = 0; o < ON; ++o) {
    const float t = tp[o * IDIM];
    m = fmaxf(m, fast_rsq(t * t * h + EPSF));
  }
  float z = 0.0f;
#pragma unroll 8
  for (int o = 0; o < ON; ++o) {
    const float t = tp[o * IDIM];
    z += __expf(fast_rsq(t * t * h + EPSF) - m);
  }
  aux[b * IDIM + i] = make_float4(h, dv / z, m, 0.0f);
}

// ---------------------------------------------------------------------------
// Kernel 2: out[b,o] = sum_i (d/Z)[b,i] * exp(s[b,o,i] - m[b,i]).
// Thread-per-o => Tt reads coalesced. The per-i aux float4 (uniform across the
// block) is staged into wave-private LDS with CDNA5 async-to-LDS, double-
// buffered with s_wait_asynccnt, then consumed via broadcast ds_read_b128.
// ---------------------------------------------------------------------------
__global__ __launch_bounds__(256) void pass2_kernel(
    const float* __restrict__ Tt, const float4* __restrict__ aux,
    float* __restrict__ out) {
  __shared__ float4 stage[8][2][32];  // [wave][buffer][i-in-chunk]; 8KB
  const int b    = blockIdx.y;
  const int o    = blockIdx.x * 256 + threadIdx.x;
  const int wave = threadIdx.x >> 5;
  const int lane = threadIdx.x & 31;
  const float4* asrc = aux + b * IDIM;

  // Issue chunk 0: each lane async-copies one 16B aux element into LDS.
  {
    const uint64_t ga = (uint64_t)(uintptr_t)(asrc + lane);
    const unsigned la = (unsigned)(uintptr_t)&stage[wave][0][lane];
    asm volatile("global_load_async_to_lds_b128 %0, %1, off"
                 :: "v"(la), "v"(ga) : "memory");
  }

  float acc = 0.0f;
  const int NCH = IDIM / 32;
  for (int ch = 0; ch < NCH; ++ch) {
    const int cur = ch & 1;
    if (ch + 1 < NCH) {
      const uint64_t ga = (uint64_t)(uintptr_t)(asrc + (ch + 1) * 32 + lane);
      const unsigned la = (unsigned)(uintptr_t)&stage[wave][cur ^ 1][lane];
      asm volatile("global_load_async_to_lds_b128 %0, %1, off"
                   :: "v"(la), "v"(ga) : "memory");
      // async loads complete in order: cnt<=1 => chunk `ch` has landed
      asm volatile("s_wait_asynccnt 0x1" ::: "memory");
    } else {
      asm volatile("s_wait_asynccnt 0x0" ::: "memory");
    }
    const float* tcol = Tt + (ch * 32) * ON + o;
#pragma unroll
    for (int j = 0; j < 32; ++j) {
      const float4 a = stage[wave][cur][j];          // broadcast ds_read_b128
      const float  t = tcol[j * ON];                 // coalesced
      const float  s = fast_rsq(t * t * a.x + EPSF);
      acc += a.y * __expf(s - a.z);
    }
  }
  out[b * ON + o] = acc;
}

// ---------------------------------------------------------------------------
extern "C" void kernel_launch(void* const* d_in, const int* in_sizes, int n_in,
                              void* d_out, int out_size, void* d_ws, size_t ws_size,
                              hipStream_t stream) {
  (void)in_sizes; (void)n_in; (void)out_size; (void)ws_size;
  const float* data        = (const float*)d_in[0];
  const float* input_x     = (const float*)d_in[1];
  const float* input_y     = (const float*)d_in[2];
  const float* output_x    = (const float*)d_in[3];
  const float* output_y    = (const float*)d_in[4];
  const float* linkage_add = (const float*)d_in[5];
  const float* linkage_mul = (const float*)d_in[6];

  float*  T   = (float*)d_ws;                 // 4 MB
  float*  Tt  = T + (size_t)ON * IDIM;        // 4 MB
  float4* aux = (float4*)(Tt + (size_t)ON * IDIM);  // 2 MB

  prep_kernel<<<dim3(IDIM / 32, ON / 32), dim3(32, 8), 0, stream>>>(
      input_x, input_y, output_x, output_y, linkage_add, linkage_mul, T, Tt);
  pass1_kernel<<<dim3(IDIM / 256, BN), 256, 0, stream>>>(data, T, aux);
  pass2_kernel<<<dim3(ON / 256, BN), 256, 0, stream>>>(Tt, aux, (float*)d_out);
}